// MultiBoxLoss_87857851007453
// MI455X (gfx1250) — compile-verified
//
#include <hip/hip_runtime.h>
#include <hip/hip_bf16.h>
#include <math.h>
#include <stdint.h>

typedef __attribute__((ext_vector_type(2))) float v2f;
typedef __attribute__((ext_vector_type(8))) float v8f;

#define D_BOX 8732
#define N_CLS 21
#define N_OBJ 16
#define NT    256
#define NEG_POS_RATIO 3
#define THRESH 0.5f
#define N_TILE ((D_BOX + NT - 1) / NT)        /* 35 tiles: 34 full + 28-row tail */
#define TILE_F (NT * N_CLS)                   /* floats per full tile = 5376     */
#define ISSUES_PER_TILE 6                     /* ceil(5376/4/256) = 6 B128/wave  */

static __device__ __forceinline__ unsigned lds_off(const void* p) {
    // addrspacecast(3->flat) keeps the LDS byte offset in the low 32 bits
    return (unsigned)(unsigned long long)p;
}

// Block-wide sum of 256 partials via V_WMMA_F32_16X16X4_F32 with an all-ones A
// fragment (D = 1^T * B + C accumulates column sums); run redundantly in all 8
// waves so EXEC is all-ones for every WMMA issue (ISA 7.12 restriction).
__device__ __forceinline__ float block_sum_wmma(float* s_red, float v) {
    const int tid  = threadIdx.x;
    const int lane = tid & 31;
    s_red[tid] = v;
    __syncthreads();
    v8f c = {};
    v2f a; a[0] = 1.0f; a[1] = 1.0f;
#pragma unroll
    for (int chunk = 0; chunk < 4; ++chunk) {
        v2f bf;
        bf[0] = s_red[chunk * 64 + lane];
        bf[1] = s_red[chunk * 64 + 32 + lane];
        c = __builtin_amdgcn_wmma_f32_16x16x4_f32(false, a, false, bf,
                                                  (short)0, c, false, false);
    }
    float r = c[0];
    r += __shfl_xor(r, 1, 32);
    r += __shfl_xor(r, 2, 32);
    r += __shfl_xor(r, 4, 32);
    r += __shfl_xor(r, 8, 32);      // every lane now holds the grand total
    __syncthreads();
    return r;
}

// Kernel 1: one block per batch item. Fused match + SmoothL1 + CE + neg-CE
// scatter, with cls_pred streamed through LDS by double-buffered
// GLOBAL_LOAD_ASYNC_TO_LDS_B128 (ASYNCcnt-pipelined).
__global__ __launch_bounds__(NT)
void k_match(const float* __restrict__ locs_pred,
             const float* __restrict__ cls_pred,
             const float* __restrict__ boxes,
             const int*   __restrict__ labels,
             const float* __restrict__ dboxes,
             float* __restrict__ ws_neg,    // [B*D]
             float* __restrict__ ws_sl1,    // [B]
             float* __restrict__ ws_cpos,   // [B]
             float* __restrict__ ws_npos)   // [B]
{
    __shared__ float s_bx[N_OBJ][4];     // object boxes, xyxy
    __shared__ float s_bc[N_OBJ][4];     // object boxes, cxcywh
    __shared__ float s_barea[N_OBJ];
    __shared__ int   s_lab[N_OBJ];
    __shared__ unsigned char s_objbest[D_BOX];
    __shared__ unsigned char s_pos[D_BOX];
    __shared__ int   s_objd[N_OBJ];
    __shared__ float s_red[NT];
    __shared__ float s_cls[2][TILE_F];   // double-buffered cls_pred tiles (2x21KB)

    const int b   = blockIdx.x;
    const int tid = threadIdx.x;

    if (tid < N_OBJ) {
        const float x0 = boxes[(b * N_OBJ + tid) * 4 + 0];
        const float y0 = boxes[(b * N_OBJ + tid) * 4 + 1];
        const float x1 = boxes[(b * N_OBJ + tid) * 4 + 2];
        const float y1 = boxes[(b * N_OBJ + tid) * 4 + 3];
        s_bx[tid][0] = x0; s_bx[tid][1] = y0; s_bx[tid][2] = x1; s_bx[tid][3] = y1;
        s_bc[tid][0] = 0.5f * (x0 + x1);
        s_bc[tid][1] = 0.5f * (y0 + y1);
        s_bc[tid][2] = x1 - x0;
        s_bc[tid][3] = y1 - y0;
        s_barea[tid] = (x1 - x0) * (y1 - y0);
        s_lab[tid]   = labels[b * N_OBJ + tid];
    }
    __syncthreads();

    // Phase 1: per-prior best object (argmax over 16 objects, first-max tie-break).
    for (int d = tid; d < D_BOX; d += NT) {
        const float4 db = ((const float4*)dboxes)[d];
        const float dx0 = db.x - 0.5f * db.z, dy0 = db.y - 0.5f * db.w;
        const float dx1 = db.x + 0.5f * db.z, dy1 = db.y + 0.5f * db.w;
        const float darea = db.z * db.w;
        float best = -1.0f; int bi = 0;
#pragma unroll
        for (int j = 0; j < N_OBJ; ++j) {
            const float lx = fmaxf(s_bx[j][0], dx0), ly = fmaxf(s_bx[j][1], dy0);
            const float rx = fminf(s_bx[j][2], dx1), ry = fminf(s_bx[j][3], dy1);
            const float iw = fmaxf(rx - lx, 0.0f),  ih = fmaxf(ry - ly, 0.0f);
            const float inter = iw * ih;
            const float iou = inter / (s_barea[j] + darea - inter);
            if (iou > best) { best = iou; bi = j; }
        }
        s_objbest[d] = (unsigned char)bi;
        s_pos[d]     = (best >= THRESH) ? 1 : 0;   // only ov_best>=0.5 matters downstream
    }
    __syncthreads();

    // Phase 2: per-object best prior (argmax over D). 8 waves x 2 objects, wave32 butterfly.
    {
        const int wave = tid >> 5, lane = tid & 31;
#pragma unroll
        for (int jj = 0; jj < 2; ++jj) {
            const int j = wave * 2 + jj;
            const float b0 = s_bx[j][0], b1 = s_bx[j][1], b2 = s_bx[j][2], b3 = s_bx[j][3];
            const float ba = s_barea[j];
            float best = -1.0f; int bd = 0;
            for (int d = lane; d < D_BOX; d += 32) {
                const float4 db = ((const float4*)dboxes)[d];
                const float dx0 = db.x - 0.5f * db.z, dy0 = db.y - 0.5f * db.w;
                const float dx1 = db.x + 0.5f * db.z, dy1 = db.y + 0.5f * db.w;
                const float lx = fmaxf(b0, dx0), ly = fmaxf(b1, dy0);
                const float rx = fminf(b2, dx1), ry = fminf(b3, dy1);
                const float iw = fmaxf(rx - lx, 0.0f), ih = fmaxf(ry - ly, 0.0f);
                const float inter = iw * ih;
                const float iou = inter / (ba + db.z * db.w - inter);
                if (iou > best) { best = iou; bd = d; }
            }
#pragma unroll
            for (int m = 16; m >= 1; m >>= 1) {
                const float ob = __shfl_xor(best, m, 32);
                const int   od = __shfl_xor(bd,   m, 32);
                if (ob > best || (ob == best && od < bd)) { best = ob; bd = od; }
            }
            if (lane == 0) s_objd[j] = bd;
        }
    }
    __syncthreads();

    // Forced-match override, sequential so duplicate indices resolve last-wins.
    if (tid == 0) {
#pragma unroll
        for (int j = 0; j < N_OBJ; ++j) {
            const int dd = s_objd[j];
            s_objbest[dd] = (unsigned char)j;
            s_pos[dd]     = 1;
        }
    }
    __syncthreads();

    // ---- Phase 3: tile pipeline over D, cls_pred async-staged into LDS ----
    const unsigned lds_base[2] = { lds_off(&s_cls[0][0]), lds_off(&s_cls[1][0]) };
    const unsigned long long gbase =
        (unsigned long long)(const void*)(cls_pred + (size_t)b * D_BOX * N_CLS);

    auto issue_tile = [&](int tile) {
        const int rows = (tile == N_TILE - 1) ? (D_BOX - tile * NT) : NT;
        const int nf4  = (rows * N_CLS) >> 2;              // rows%4==0 -> exact
        const unsigned tile_byte = (unsigned)(tile * TILE_F) * 4u;
        const unsigned lb = lds_base[tile & 1];
#pragma unroll
        for (int k = 0; k < ISSUES_PER_TILE; ++k) {
            int i = tid + k * NT;
            if (i >= nf4) i = nf4 - 1;                     // clamp: fixed 6 issues/wave
            const unsigned voff = tile_byte + (unsigned)i * 16u;
            const unsigned ldso = lb + (unsigned)i * 16u;
            asm volatile("global_load_async_to_lds_b128 %0, %1, %2"
                         :: "v"(ldso), "v"(voff), "s"(gbase) : "memory");
        }
    };

    issue_tile(0);

    float sl1_acc = 0.0f, cpos_acc = 0.0f, npos_acc = 0.0f;
    for (int tile = 0; tile < N_TILE; ++tile) {
        if (tile + 1 < N_TILE) {
            issue_tile(tile + 1);
            // in-order completion: <=6 outstanding == only next tile's loads remain
            asm volatile("s_wait_asynccnt 6" ::: "memory");
        } else {
            asm volatile("s_wait_asynccnt 0" ::: "memory");
        }
        __syncthreads();                                    // tile visible block-wide

        const int d = tile * NT + tid;
        if (d < D_BOX) {
            const int  j   = s_objbest[d];
            const bool pos = s_pos[d] != 0;
            const int  lab = pos ? s_lab[j] : 0;

            if (pos) {
                const float4 db = ((const float4*)dboxes)[d];
                const float gcx = (s_bc[j][0] - db.x) / (db.z * 0.1f);
                const float gcy = (s_bc[j][1] - db.y) / (db.w * 0.1f);
                const float gw  = __logf(s_bc[j][2] / db.z) * 5.0f;
                const float gh  = __logf(s_bc[j][3] / db.w) * 5.0f;
                const float4 lp = ((const float4*)locs_pred)[(size_t)b * D_BOX + d];
                const float e0 = lp.x - gcx, e1 = lp.y - gcy, e2 = lp.z - gw, e3 = lp.w - gh;
                float s = 0.0f, ad;
                ad = fabsf(e0); s += (ad < 1.0f) ? 0.5f * e0 * e0 : ad - 0.5f;
                ad = fabsf(e1); s += (ad < 1.0f) ? 0.5f * e1 * e1 : ad - 0.5f;
                ad = fabsf(e2); s += (ad < 1.0f) ? 0.5f * e2 * e2 : ad - 0.5f;
                ad = fabsf(e3); s += (ad < 1.0f) ? 0.5f * e3 * e3 : ad - 0.5f;
                sl1_acc  += s;
                npos_acc += 1.0f;
            }

            // CE = logsumexp(x) - x[lab]; row stride 21 words is coprime with
            // the bank count -> conflict-free LDS reads.
            const float* xr = &s_cls[tile & 1][tid * N_CLS];
            float x[N_CLS];
            float m = -1e30f, xlab = 0.0f;
#pragma unroll
            for (int c = 0; c < N_CLS; ++c) {
                const float v = xr[c];
                x[c] = v;
                m = fmaxf(m, v);
                if (c == lab) xlab = v;        // avoid dynamic register indexing
            }
            float se = 0.0f;
#pragma unroll
            for (int c = 0; c < N_CLS; ++c) se += __expf(x[c] - m);
            float ce = m + __logf(se) - xlab;
            if (pos) { cpos_acc += ce; ce = 0.0f; }
            ws_neg[(size_t)b * D_BOX + d] = ce;
        }
        __syncthreads();                        // everyone done before buffer reuse
    }

    const float sl1_tot  = block_sum_wmma(s_red, sl1_acc);
    const float cpos_tot = block_sum_wmma(s_red, cpos_acc);
    const float npos_tot = block_sum_wmma(s_red, npos_acc);
    if (tid == 0) {
        ws_sl1[b]  = sl1_tot;
        ws_cpos[b] = cpos_tot;
        ws_npos[b] = npos_tot;
    }
}

// Kernel 2: hard-negative mining = sum of top-K negatives via radix-select on
// float bit patterns. Row is bulk-filled into LDS with async B128 loads.
__global__ __launch_bounds__(NT)
void k_hardneg(const float* __restrict__ ws_neg,
               const float* __restrict__ ws_npos,
               float* __restrict__ ws_hard)
{
    __shared__ unsigned int s_v[D_BOX];     // 8732 floats = 2183 B128 transfers exactly
    __shared__ int   s_cnt[NT];
    __shared__ float s_f[NT];

    const int b   = blockIdx.x;
    const int tid = threadIdx.x;

    {
        const unsigned lb = lds_off(&s_v[0]);
        const unsigned long long gbase =
            (unsigned long long)(const void*)(ws_neg + (size_t)b * D_BOX);
        const int nf4 = D_BOX / 4;          // 2183
#pragma unroll
        for (int k = 0; k < 9; ++k) {       // ceil(2183/256) = 9 issues/wave
            int i = tid + k * NT;
            if (i >= nf4) i = nf4 - 1;
            const unsigned off = (unsigned)i * 16u;
            asm volatile("global_load_async_to_lds_b128 %0, %1, %2"
                         :: "v"(lb + off), "v"(off), "s"(gbase) : "memory");
        }
        asm volatile("s_wait_asynccnt 0" ::: "memory");
    }

    const int npos = (int)(ws_npos[b] + 0.5f);
    int K = NEG_POS_RATIO * npos;
    if (K > D_BOX) K = D_BOX;
    __syncthreads();

    if (K <= 0) {
        if (tid == 0) ws_hard[b] = 0.0f;
        return;
    }

    unsigned int X = 0u;                      // K-th largest bit pattern
    for (int bit = 30; bit >= 0; --bit) {     // sign bit always 0 (CE >= 0)
        const unsigned int trial = X | (1u << bit);
        int cnt = 0;
        for (int d = tid; d < D_BOX; d += NT) cnt += (s_v[d] >= trial) ? 1 : 0;
        s_cnt[tid] = cnt;
        __syncthreads();
        for (int s = NT / 2; s > 0; s >>= 1) {
            if (tid < s) s_cnt[tid] += s_cnt[tid + s];
            __syncthreads();
        }
        if (s_cnt[0] >= K) X = trial;         // uniform decision in all threads
        __syncthreads();
    }

    float sumGT = 0.0f; int cntGT = 0;
    for (int d = tid; d < D_BOX; d += NT) {
        const unsigned int v = s_v[d];
        if (v > X) { sumGT += __uint_as_float(v); cntGT++; }
    }
    s_cnt[tid] = cntGT; s_f[tid] = sumGT;
    __syncthreads();
    for (int s = NT / 2; s > 0; s >>= 1) {
        if (tid < s) { s_cnt[tid] += s_cnt[tid + s]; s_f[tid] += s_f[tid + s]; }
        __syncthreads();
    }
    if (tid == 0)
        ws_hard[b] = s_f[0] + (float)(K - s_cnt[0]) * __uint_as_float(X);
}

// Kernel 3: combine per-batch partials into the scalar loss.
__global__ __launch_bounds__(NT)
void k_finish(const float* __restrict__ ws_sl1,
              const float* __restrict__ ws_cpos,
              const float* __restrict__ ws_npos,
              const float* __restrict__ ws_hard,
              float* __restrict__ out, int B)
{
    __shared__ float s_red[NT];
    const int tid = threadIdx.x;
    float a = 0.0f, c = 0.0f, n = 0.0f, h = 0.0f;
    for (int i = tid; i < B; i += NT) {
        a += ws_sl1[i]; c += ws_cpos[i]; n += ws_npos[i]; h += ws_hard[i];
    }
    const float at = block_sum_wmma(s_red, a);
    const float ct = block_sum_wmma(s_red, c);
    const float nt = block_sum_wmma(s_red, n);
    const float ht = block_sum_wmma(s_red, h);
    if (tid == 0)
        out[0] = at / (4.0f * nt) + (ht + ct) / nt;
}

extern "C" void kernel_launch(void* const* d_in, const int* in_sizes, int n_in,
                              void* d_out, int out_size, void* d_ws, size_t ws_size,
                              hipStream_t stream) {
    const float* locs_pred = (const float*)d_in[0];
    const float* cls_pred  = (const float*)d_in[1];
    const float* boxes     = (const float*)d_in[2];
    const int*   labels    = (const int*)d_in[3];
    const float* dboxes    = (const float*)d_in[4];

    const int B = in_sizes[3] / N_OBJ;   // 128

    float* wsf     = (float*)d_ws;
    float* ws_neg  = wsf;                              // B*D
    float* ws_sl1  = wsf + (size_t)B * D_BOX;          // B
    float* ws_cpos = ws_sl1 + B;                       // B
    float* ws_npos = ws_cpos + B;                      // B
    float* ws_hard = ws_npos + B;                      // B

    k_match  <<<B, NT, 0, stream>>>(locs_pred, cls_pred, boxes, labels, dboxes,
                                    ws_neg, ws_sl1, ws_cpos, ws_npos);
    k_hardneg<<<B, NT, 0, stream>>>(ws_neg, ws_npos, ws_hard);
    k_finish <<<1, NT, 0, stream>>>(ws_sl1, ws_cpos, ws_npos, ws_hard,
                                    (float*)d_out, B);
}